// BiATT_41601053229597
// MI455X (gfx1250) — compile-verified
//
#include <hip/hip_runtime.h>

// BiATT on MI455X (gfx1250).
// Algebraic simplification: softmax over axis=1 of a [N,1] tensor == 1, so
//   cf = atoms @ (sum of 4 DxD blocks of Wcc) + bcc
//   pf = amino @ (sum of 4 DxD blocks of Wcp) + bcp
// Everything else in the reference is dead code. The problem is HBM-bound:
// ~60 MB total traffic -> ~2.6 us at 23.3 TB/s; 6.4 GFLOP fp32 via
// v_wmma_f32_16x16x4_f32 is free at that intensity.
//
// Round-1 asm showed v_dual_mov repacking of B fragments (K / K+1 rows were
// SW_STRIDE apart in LDS). This version stores W pair-interleaved
// ({W[2p][n], W[2p+1][n]} contiguous) so a B fragment is one aligned
// ds_load_b64 straight into the even VGPR pair WMMA wants. Pair-row stride
// of 80 float2s puts adjacent pair-rows +32 banks apart, so the two
// half-waves read complementary bank sets (conflict-free).

typedef __attribute__((ext_vector_type(2))) float v2f;
typedef __attribute__((ext_vector_type(4))) float v4f;
typedef __attribute__((ext_vector_type(8))) float v8f;

#define DDIM 512
#define NROWS 6144
#define BLK_M 128
#define BLK_N 64
#define BLK_K 32
#define SA_STRIDE 34        // floats; half-waves land on complementary banks
#define SW2_STRIDE 80       // float2 units; +1 pair-row == +32 banks

// Fold Wcc/Wcp [4D, D] -> [D, D] block sums (the only part of the weights
// that survives the softmax degeneracy).
__global__ __launch_bounds__(256)
void biatt_fold_weights(const float* __restrict__ Wcc,
                        const float* __restrict__ Wcp,
                        float* __restrict__ WsumCC,
                        float* __restrict__ WsumCP) {
    int idx = blockIdx.x * blockDim.x + threadIdx.x;
    if (idx >= DDIM * DDIM) return;
    const int S = DDIM * DDIM;
    WsumCC[idx] = Wcc[idx] + Wcc[idx + S] + Wcc[idx + 2 * S] + Wcc[idx + 3 * S];
    WsumCP[idx] = Wcp[idx] + Wcp[idx + S] + Wcp[idx + 2 * S] + Wcp[idx + 3 * S];
}

// C[M=6144, N=512] = A[M, 512] @ W[512, N] + bias[N], fp32 WMMA.
// 8 waves / block; wave w computes M-tile w (16 rows) x 4 N-tiles (64 cols).
// blockIdx.z selects (atoms, WsumCC, bcc, cf) vs (amino, WsumCP, bcp, pf).
__global__ __launch_bounds__(256)
void biatt_gemm(const float* __restrict__ A0, const float* __restrict__ A1,
                const float* __restrict__ W0, const float* __restrict__ W1,
                const float* __restrict__ b0, const float* __restrict__ b1,
                float* __restrict__ C0, float* __restrict__ C1) {
    const float* __restrict__ A    = blockIdx.z ? A1 : A0;
    const float* __restrict__ W    = blockIdx.z ? W1 : W0;
    const float* __restrict__ bias = blockIdx.z ? b1 : b0;
    float* __restrict__ C          = blockIdx.z ? C1 : C0;

    __shared__ __align__(16) float sA[BLK_M * SA_STRIDE];        // ~17 KB
    __shared__ __align__(16) float sW[(BLK_K / 2) * SW2_STRIDE * 2]; // 10 KB

    const int tid    = threadIdx.x;
    const int lane   = tid & 31;
    const int wave   = tid >> 5;      // 0..7 -> M-tile within block
    const int laneLo = lane & 15;     // N (for B/C/D) or M (for A) index
    const int laneHi = lane >> 4;     // selects K pair (A/B) / M+8 (C/D)

    const int blockM = blockIdx.x * BLK_M;
    const int blockN = blockIdx.y * BLK_N;

    // per-thread staging coordinates
    const int aR  = tid >> 3;            // 0..31 (A tile row group)
    const int aC4 = (tid & 7) * 4;       // 0..28
    const int wP  = tid >> 4;            // 0..15 (W pair-row: k = 2p, 2p+1)
    const int wC4 = (tid & 15) * 4;      // 0..60

    v8f acc[4] = {};                     // four 16x16 f32 accumulators

    for (int kb = 0; kb < DDIM; kb += BLK_K) {
        // --- Stage A tile 128x32: 4x coalesced float4 per thread ---
        #pragma unroll
        for (int it = 0; it < 4; ++it) {
            int row = aR + it * 32;
            v4f v = *(const v4f*)(A + (size_t)(blockM + row) * DDIM + kb + aC4);
            float* s = &sA[row * SA_STRIDE + aC4];
            s[0] = v.x; s[1] = v.y; s[2] = v.z; s[3] = v.w;
        }
        // --- Stage W tile 32x64, pair-interleaved: sW2[p][n] = {W[2p][n], W[2p+1][n]} ---
        {
            const float* g0 = W + (size_t)(kb + 2 * wP    ) * DDIM + blockN + wC4;
            const float* g1 = W + (size_t)(kb + 2 * wP + 1) * DDIM + blockN + wC4;
            v4f v0 = *(const v4f*)g0;
            v4f v1 = *(const v4f*)g1;
            v4f lo = {v0.x, v1.x, v0.y, v1.y};
            v4f hi = {v0.z, v1.z, v0.w, v1.w};
            float* s = &sW[(wP * SW2_STRIDE + wC4) * 2];
            *(v4f*)(s)     = lo;
            *(v4f*)(s + 4) = hi;
        }

        // --- Prefetch next K-tile into cache (global_prefetch_b8) ---
        if (kb + BLK_K < DDIM) {
            __builtin_prefetch(A + (size_t)(blockM + aR) * DDIM + kb + BLK_K + aC4, 0, 3);
            __builtin_prefetch(W + (size_t)(kb + BLK_K + 2 * wP) * DDIM + blockN + wC4, 0, 3);
        }
        __syncthreads();

        const int m0 = wave * 16;
        #pragma unroll
        for (int ks = 0; ks < BLK_K / 4; ++ks) {
            // fp32 A-frag 16x4: lanes 0-15 hold K0/K1, lanes 16-31 hold K2/K3.
            const int kk = ks * 4 + laneHi * 2;
            v2f a = *(const v2f*)&sA[(m0 + laneLo) * SA_STRIDE + kk];
            const int p = ks * 2 + laneHi;   // pair-row holding {K=kk, K=kk+1}
            #pragma unroll
            for (int nt = 0; nt < 4; ++nt) {
                // fp32 B-frag 4x16: single aligned b64 -> adjacent VGPR pair.
                v2f b = *(const v2f*)&sW[(p * SW2_STRIDE + nt * 16 + laneLo) * 2];
                acc[nt] = __builtin_amdgcn_wmma_f32_16x16x4_f32(
                    /*neg_a=*/false, a, /*neg_b=*/false, b,
                    /*c_mod=*/(short)0, acc[nt],
                    /*reuse_a=*/false, /*reuse_b=*/false);
            }
        }
        __syncthreads();
    }

    // Epilogue: C/D layout -> VGPR r, lanes 0-15: (M=r, N=lane),
    // lanes 16-31: (M=r+8, N=lane-16). Add bias, store fp32.
    const int mBase = blockM + wave * 16 + laneHi * 8;
    #pragma unroll
    for (int nt = 0; nt < 4; ++nt) {
        int col  = blockN + nt * 16 + laneLo;
        float bv = bias[col];
        #pragma unroll
        for (int r = 0; r < 8; ++r) {
            C[(size_t)(mBase + r) * DDIM + col] = acc[nt][r] + bv;
        }
    }
}

extern "C" void kernel_launch(void* const* d_in, const int* in_sizes, int n_in,
                              void* d_out, int out_size, void* d_ws, size_t ws_size,
                              hipStream_t stream) {
    // setup_inputs() order:
    //  0 atoms_vector  1 amino_vector  2 U  3 Wp2c  4 bp2c  5 Wc2p  6 bc2p
    //  7 Whc  8 bhc  9 Whp 10 bhp 11 Wac 12 bac 13 Wap 14 bap
    // 15 Wcc 16 bcc 17 Wcp 18 bcp
    const float* atoms = (const float*)d_in[0];
    const float* amino = (const float*)d_in[1];
    const float* Wcc   = (const float*)d_in[15];
    const float* bcc   = (const float*)d_in[16];
    const float* Wcp   = (const float*)d_in[17];
    const float* bcp   = (const float*)d_in[18];

    float* cf = (float*)d_out;                       // [6144, 512]
    float* pf = cf + (size_t)NROWS * DDIM;           // [6144, 512]

    float* WsumCC = (float*)d_ws;                    // [512, 512]
    float* WsumCP = WsumCC + DDIM * DDIM;            // [512, 512]

    biatt_fold_weights<<<(DDIM * DDIM + 255) / 256, 256, 0, stream>>>(
        Wcc, Wcp, WsumCC, WsumCP);

    dim3 grid(NROWS / BLK_M, DDIM / BLK_N, 2);       // 48 x 8 x 2 blocks
    biatt_gemm<<<grid, 256, 0, stream>>>(
        atoms, amino, WsumCC, WsumCP, bcc, bcp, cf, pf);
}